// ComplexDotLayer_43946105373353
// MI455X (gfx1250) — compile-verified
//
#include <hip/hip_runtime.h>

typedef __attribute__((ext_vector_type(2))) float v2f;
typedef __attribute__((ext_vector_type(8))) float v8f;

#define BATCH 32
#define HGT   32
#define WID   32
#define CIN   128
#define COUT  128
#define BN_EPS 1e-3f

// One workgroup per spatial position p = h*W + w (1024 WGs).
// 8 wave32s per WG; wave -> one 16-wide cout tile x BOTH 16-row batch tiles
// (B fragments reused across the two M tiles: 1.5 vmem loads per WMMA).
// Eight f32 WMMA accumulator chains per wave over K=CIN in steps of 4.
__global__ __launch_bounds__(256) void ComplexDotLayer_kernel(
    const float* __restrict__ xr, const float* __restrict__ xi,
    const float* __restrict__ wr, const float* __restrict__ wi,
    const float* __restrict__ gr, const float* __restrict__ br,
    const float* __restrict__ mr, const float* __restrict__ vrr,
    const float* __restrict__ gi, const float* __restrict__ bi,
    const float* __restrict__ mi, const float* __restrict__ vii,
    float* __restrict__ out)
{
    const int p    = blockIdx.x;            // spatial position, 0..1023
    const int nt   = threadIdx.x >> 5;      // cout tile 0..7
    const int lane = threadIdx.x & 31;

    const int l15   = lane & 15;
    const int half  = lane >> 4;            // 0 or 1 (half-wave)
    const int khalf = half * 2;             // K offset per ISA A/B layout
    const int ncol  = nt * 16 + l15;        // out channel for B/C/D fragment

    const size_t xbstride = (size_t)HGT * WID * CIN;   // batch stride in x
    // A fragment rows for batch tiles mt=0 (rows 0..15) and mt=1 (rows 16..31)
    const float* xr_row0 = xr + (size_t)p * CIN + (size_t)l15 * xbstride;
    const float* xi_row0 = xi + (size_t)p * CIN + (size_t)l15 * xbstride;
    const float* xr_row1 = xr_row0 + 16 * xbstride;
    const float* xi_row1 = xi_row0 + 16 * xbstride;
    const float* wr_p    = wr + (size_t)p * CIN * COUT;
    const float* wi_p    = wi + (size_t)p * CIN * COUT;

    v8f acc_rr0 = {}, acc_ii0 = {}, acc_ri0 = {}, acc_ir0 = {};
    v8f acc_rr1 = {}, acc_ii1 = {}, acc_ri1 = {}, acc_ir1 = {};

    #pragma unroll 2
    for (int k0 = 0; k0 < CIN; k0 += 4) {
        const int ka = k0 + khalf;
        // A fragments: lane holds X[row][ka], X[row][ka+1] (8B contiguous)
        v2f a_r0 = *(const v2f*)(xr_row0 + ka);
        v2f a_i0 = *(const v2f*)(xi_row0 + ka);
        v2f a_r1 = *(const v2f*)(xr_row1 + ka);
        v2f a_i1 = *(const v2f*)(xi_row1 + ka);
        // B fragments: lane holds W[ka][ncol], W[ka+1][ncol] — shared by both
        // batch tiles.
        v2f b_r, b_i;
        b_r.x = wr_p[(size_t)ka * COUT + ncol];
        b_r.y = wr_p[(size_t)(ka + 1) * COUT + ncol];
        b_i.x = wi_p[(size_t)ka * COUT + ncol];
        b_i.y = wi_p[(size_t)(ka + 1) * COUT + ncol];

        acc_rr0 = __builtin_amdgcn_wmma_f32_16x16x4_f32(
            false, a_r0, false, b_r, (short)0, acc_rr0, false, false);
        acc_ri0 = __builtin_amdgcn_wmma_f32_16x16x4_f32(
            false, a_r0, false, b_i, (short)0, acc_ri0, false, false);
        acc_ii0 = __builtin_amdgcn_wmma_f32_16x16x4_f32(
            false, a_i0, false, b_i, (short)0, acc_ii0, false, false);
        acc_ir0 = __builtin_amdgcn_wmma_f32_16x16x4_f32(
            false, a_i0, false, b_r, (short)0, acc_ir0, false, false);
        acc_rr1 = __builtin_amdgcn_wmma_f32_16x16x4_f32(
            false, a_r1, false, b_r, (short)0, acc_rr1, false, false);
        acc_ri1 = __builtin_amdgcn_wmma_f32_16x16x4_f32(
            false, a_r1, false, b_i, (short)0, acc_ri1, false, false);
        acc_ii1 = __builtin_amdgcn_wmma_f32_16x16x4_f32(
            false, a_i1, false, b_i, (short)0, acc_ii1, false, false);
        acc_ir1 = __builtin_amdgcn_wmma_f32_16x16x4_f32(
            false, a_i1, false, b_r, (short)0, acc_ir1, false, false);
    }

    // BatchNorm coefficients for this lane's channel (one channel per lane).
    const float inv_r = gr[ncol] * rsqrtf(vrr[ncol] + BN_EPS);
    const float add_r = br[ncol] - mr[ncol] * inv_r;
    const float inv_i = gi[ncol] * rsqrtf(vii[ncol] + BN_EPS);
    const float add_i = bi[ncol] - mi[ncol] * inv_i;

    float* out_r = out;
    float* out_i = out + (size_t)BATCH * HGT * WID * COUT;

    // D layout: VGPR r, lane -> M = r + 8*half (+ tile), N = l15.
    const int mbase = 8 * half;
    #pragma unroll
    for (int r = 0; r < 8; ++r) {
        const int b0 = mbase + r;          // batch tile 0
        const int b1 = b0 + 16;            // batch tile 1
        const size_t off0 = (size_t)b0 * ((size_t)HGT * WID * COUT)
                          + (size_t)p * COUT + ncol;
        const size_t off1 = (size_t)b1 * ((size_t)HGT * WID * COUT)
                          + (size_t)p * COUT + ncol;
        out_r[off0] = (acc_rr0[r] + acc_ii0[r]) * inv_r + add_r;
        out_i[off0] = (acc_ri0[r] - acc_ir0[r]) * inv_i + add_i;
        out_r[off1] = (acc_rr1[r] + acc_ii1[r]) * inv_r + add_r;
        out_i[off1] = (acc_ri1[r] - acc_ir1[r]) * inv_i + add_i;
    }
}

extern "C" void kernel_launch(void* const* d_in, const int* in_sizes, int n_in,
                              void* d_out, int out_size, void* d_ws, size_t ws_size,
                              hipStream_t stream) {
    (void)in_sizes; (void)n_in; (void)out_size; (void)d_ws; (void)ws_size;
    const float* xr  = (const float*)d_in[0];
    const float* xi  = (const float*)d_in[1];
    const float* wr  = (const float*)d_in[2];
    const float* wi  = (const float*)d_in[3];
    const float* gr  = (const float*)d_in[4];
    const float* br  = (const float*)d_in[5];
    const float* mr  = (const float*)d_in[6];
    const float* vrr = (const float*)d_in[7];
    const float* gi  = (const float*)d_in[8];
    const float* bi  = (const float*)d_in[9];
    const float* mi  = (const float*)d_in[10];
    const float* vii = (const float*)d_in[11];
    float* out = (float*)d_out;

    dim3 grid(HGT * WID);   // 1024 positions
    dim3 block(256);        // 8 wave32s
    ComplexDotLayer_kernel<<<grid, block, 0, stream>>>(
        xr, xi, wr, wi, gr, br, mr, vrr, gi, bi, mi, vii, out);
}